// DSS_22505628631650
// MI455X (gfx1250) — compile-verified
//
#include <hip/hip_runtime.h>
#include <math.h>

#define B_  2
#define N_  8192
#define H_  256
#define W_  256
#define R2_ 0.01f      // RADIUS^2
#define EPS_ 1e-8f

typedef __attribute__((ext_vector_type(2))) float v2f;
typedef __attribute__((ext_vector_type(8))) float v8f;

// ---------------------------------------------------------------------------
// zero the image accumulators in workspace
// ---------------------------------------------------------------------------
__global__ __launch_bounds__(256) void dss_zero(float* __restrict__ p, int n) {
    int i = blockIdx.x * 256 + threadIdx.x;
    if (i < n) p[i] = 0.0f;
}

// ---------------------------------------------------------------------------
// KNN-gaussian density via V_WMMA_F32_16X16X4_F32.
//   A row i  = [-2x_i, -2y_i, -2z_i, 1]          (16x4 f32, 2 VGPRs)
//   B col j  = [ x_j,   y_j,   z_j, |p_j|^2 ]    (4x16 f32, 2 VGPRs)
//   C(r,c)   = |p_i|^2  (broadcast per row)      (16x16 f32, 8 VGPRs)
//   D        = |p_i|^2 + |p_j|^2 - 2 p_i.p_j = d^2   directly from the MMA.
// weight = exp(-d2/(4 r^2)) gated by d2 <= r^2; self-exclusion handled only in
// the (wave-uniform) diagonal tile. One wave owns one 16-row i-tile, loops all
// 512 j-tiles unrolled x2 -> two independent WMMAs in flight. Density stored
// with plain writes (no atomics). Grid = exactly B*N/16 waves (full EXEC).
// ---------------------------------------------------------------------------
__global__ __launch_bounds__(256) void dss_density_wmma(
        const float* __restrict__ points, float* __restrict__ dens) {
    const int wave  = (blockIdx.x * 256 + (int)threadIdx.x) >> 5;  // 0..1023
    const int lane  = threadIdx.x & 31;
    const int half  = lane >> 4;       // 0: K0/K1 + rows 0..7 ; 1: K2/K3 + rows 8..15
    const int col   = lane & 15;
    const int b     = wave >> 9;       // 512 i-tiles per batch
    const int itile = wave & 511;
    const int iBase = itile << 4;
    const float* __restrict__ Pb = points + (size_t)b * N_ * 3;

    // A fragment (loop invariant across all j-tiles)
    const float ax = Pb[3 * (iBase + col) + 0];
    const float ay = Pb[3 * (iBase + col) + 1];
    const float az = Pb[3 * (iBase + col) + 2];
    v2f afrag;
    afrag.x = half ? (-2.0f * az) : (-2.0f * ax);   // V0: K=0 | K=2
    afrag.y = half ? ( 1.0f     ) : (-2.0f * ay);   // V1: K=1 | K=3

    // C fragment: |p_i|^2 for the 8 rows this lane's C/D registers map to.
    // C element (row, col) = sq_i[row], identical across the 16 columns,
    // which is exactly one scalar per (lane, v) in the 16x16 f32 C layout.
    v8f cfrag;
#pragma unroll
    for (int v = 0; v < 8; ++v) {
        const int ip = iBase + half * 8 + v;
        const float xx = Pb[3 * ip + 0], yy = Pb[3 * ip + 1], zz = Pb[3 * ip + 2];
        cfrag[v] = xx * xx + yy * yy + zz * zz;
    }

    float acc[8];
#pragma unroll
    for (int v = 0; v < 8; ++v) acc[v] = 0.0f;

    // epilogue for one 16x16 tile of d^2 values
    auto accumulate = [&](const v8f& d, int jt) {
        if (jt != itile) {                    // wave-uniform scalar branch
#pragma unroll
            for (int v = 0; v < 8; ++v) {
                const float d2 = fmaxf(d[v], 0.0f);
                const float w  = __expf(-25.0f * d2);      // 1/(4 r^2) = 25
                acc[v] += (d2 <= R2_) ? w : 0.0f;
            }
        } else {                              // diagonal tile: exclude self
#pragma unroll
            for (int v = 0; v < 8; ++v) {
                const float d2 = fmaxf(d[v], 0.0f);
                const float w  = __expf(-25.0f * d2);
                const bool ok  = (d2 <= R2_) && (col != half * 8 + v);
                acc[v] += ok ? w : 0.0f;
            }
        }
    };

    for (int jt = 0; jt < N_ / 16; jt += 2) {
        const int j0 = (jt << 4) + col;
        const int j1 = j0 + 16;
        // pull future tiles toward the WGP (lowers to global_prefetch_b8)
        if (jt + 8 < N_ / 16) __builtin_prefetch(&Pb[3 * (j0 + 128)], 0, 1);

        const float qx0 = Pb[3 * j0 + 0], qy0 = Pb[3 * j0 + 1], qz0 = Pb[3 * j0 + 2];
        const float qx1 = Pb[3 * j1 + 0], qy1 = Pb[3 * j1 + 1], qz1 = Pb[3 * j1 + 2];
        const float sq0 = qx0 * qx0 + qy0 * qy0 + qz0 * qz0;
        const float sq1 = qx1 * qx1 + qy1 * qy1 + qz1 * qz1;

        v2f b0, b1;
        b0.x = half ? qy0 : qx0;   // V0: K=0 (lanes 0-15) | K=1 (lanes 16-31)
        b0.y = half ? sq0 : qz0;   // V1: K=2 (lanes 0-15) | K=3 (lanes 16-31)
        b1.x = half ? qy1 : qx1;
        b1.y = half ? sq1 : qz1;

        // two independent WMMAs in flight; epilogue co-executes (XDL = TRANS)
        v8f d0 = __builtin_amdgcn_wmma_f32_16x16x4_f32(
            false, afrag, false, b0, (short)0, cfrag, false, false);
        v8f d1 = __builtin_amdgcn_wmma_f32_16x16x4_f32(
            false, afrag, false, b1, (short)0, cfrag, false, false);

        accumulate(d0, jt);
        accumulate(d1, jt + 1);
    }

    // reduce over the 16 columns held within each half-wave
#pragma unroll
    for (int v = 0; v < 8; ++v) {
        float s = acc[v];
        s += __shfl_xor(s, 1, 16);
        s += __shfl_xor(s, 2, 16);
        s += __shfl_xor(s, 4, 16);
        s += __shfl_xor(s, 8, 16);
        acc[v] = s;
    }
    if (col == 0) {
#pragma unroll
        for (int v = 0; v < 8; ++v)
            dens[b * N_ + iBase + half * 8 + v] = acc[v];
    }
}

// ---------------------------------------------------------------------------
// Elliptical gaussian splat: one thread per point, 9x9 fixed footprint,
// float atomics into img_c [BHW,3] / img_w [BHW] in workspace.
// ---------------------------------------------------------------------------
__global__ __launch_bounds__(256) void dss_splat(
        const float* __restrict__ proj, const float* __restrict__ ell,
        const float* __restrict__ colors, const float* __restrict__ dens,
        float* __restrict__ img_c, float* __restrict__ img_w) {
    const int t = blockIdx.x * 256 + threadIdx.x;
    if (t >= B_ * N_) return;
    const int b = t / N_;

    const float ea = ell[3 * t + 0];
    const float eb = ell[3 * t + 1];
    const float ec = ell[3 * t + 2];
    const float x  = proj[2 * t + 0];
    const float y  = proj[2 * t + 1];
    const float cr = colors[3 * t + 0];
    const float cg = colors[3 * t + 1];
    const float cb = colors[3 * t + 2];
    const float de = dens[t];

    const float den = 4.0f * ea * eb - ec * ec;
    const float xE  = fminf(sqrtf(4.0f * eb / den), 15.0f);   // CUTOFF = 1
    const float yE  = fminf(sqrtf(4.0f * ea / den), 15.0f);
    const float x0  = floorf(x);
    const float y0  = floorf(y);

    for (int oyi = -4; oyi <= 4; ++oyi) {
        for (int oxi = -4; oxi <= 4; ++oxi) {
            const float pxf = x0 + (float)oxi;
            const float pyf = y0 + (float)oyi;
            const float dx = pxf - x;
            const float dy = pyf - y;
            const float Q  = ea * dx * dx + eb * dy * dy + ec * dx * dy;
            const bool inb  = (pxf >= 0.0f) && (pxf < (float)W_) &&
                              (pyf >= 0.0f) && (pyf < (float)H_);
            const bool bbox = (fabsf(dx) <= xE + 1.0f) && (fabsf(dy) <= yE + 1.0f);
            if (inb && bbox && (Q <= 1.0f)) {
                const float w = __expf(-0.5f * Q) * de;
                const int pxi = min(max((int)pxf, 0), W_ - 1);
                const int pyi = min(max((int)pyf, 0), H_ - 1);
                const int idx = (b * H_ + pyi) * W_ + pxi;
                atomicAdd(&img_w[idx], w);
                atomicAdd(&img_c[3 * idx + 0], w * cr);
                atomicAdd(&img_c[3 * idx + 1], w * cg);
                atomicAdd(&img_c[3 * idx + 2], w * cb);
            }
        }
    }
}

// ---------------------------------------------------------------------------
// image = img_c / (img_w + eps)
// ---------------------------------------------------------------------------
__global__ __launch_bounds__(256) void dss_normalize(
        const float* __restrict__ img_c, const float* __restrict__ img_w,
        float* __restrict__ out) {
    const int p = blockIdx.x * 256 + threadIdx.x;
    if (p >= B_ * H_ * W_) return;
    const float inv = 1.0f / (img_w[p] + EPS_);
    out[3 * p + 0] = img_c[3 * p + 0] * inv;
    out[3 * p + 1] = img_c[3 * p + 1] * inv;
    out[3 * p + 2] = img_c[3 * p + 2] * inv;
}

// ---------------------------------------------------------------------------
extern "C" void kernel_launch(void* const* d_in, const int* in_sizes, int n_in,
                              void* d_out, int out_size, void* d_ws, size_t ws_size,
                              hipStream_t stream) {
    const float* points = (const float*)d_in[0];   // [B,N,3]
    const float* proj   = (const float*)d_in[1];   // [B,N,2]
    const float* ell    = (const float*)d_in[2];   // [B,N,3]
    const float* colors = (const float*)d_in[3];   // [B,N,3]
    float* out = (float*)d_out;                    // [B,H,W,3]

    // workspace layout: density | img_c | img_w
    float* ws    = (float*)d_ws;
    float* dens  = ws;                                   // B*N
    float* img_c = ws + (size_t)B_ * N_;                 // B*H*W*3
    float* img_w = img_c + (size_t)B_ * H_ * W_ * 3;     // B*H*W

    const int nz = B_ * H_ * W_ * 4;                     // img_c + img_w contiguous
    dss_zero<<<(nz + 255) / 256, 256, 0, stream>>>(img_c, nz);

    // B*N/16 = 1024 waves, 8 waves / 256-thread block -> 128 blocks (exact fit)
    dss_density_wmma<<<(B_ * (N_ / 16) * 32) / 256, 256, 0, stream>>>(points, dens);

    dss_splat<<<(B_ * N_ + 255) / 256, 256, 0, stream>>>(proj, ell, colors, dens,
                                                         img_c, img_w);

    dss_normalize<<<(B_ * H_ * W_ + 255) / 256, 256, 0, stream>>>(img_c, img_w, out);
}